// Model_14259291422784
// MI455X (gfx1250) — compile-verified
//
#include <hip/hip_runtime.h>
#include <math.h>
#include <stdint.h>

typedef __attribute__((ext_vector_type(2))) float v2f;
typedef __attribute__((ext_vector_type(8))) float v8f;

#define N_PTS 32768
#define M_PTS 8192
#define C_DIM 128
#define TILE_M 2048   // pos2 LDS tile
#define EPS_BN 1e-5f
#define EPS_D  1e-8f

// ---------------------------------------------------------------------------
// CDNA5 async global->LDS copies (ASYNCcnt-tracked, no VGPR round-trip).
// VDST operand carries the 32-bit LDS byte offset (low 32 bits of the flat
// address of a __shared__ object), VADDR the 64-bit global address.
// ---------------------------------------------------------------------------
__device__ __forceinline__ void async_copy_b64(void* lds_dst, const void* src) {
    const uint32_t lds_off = (uint32_t)(uintptr_t)lds_dst;
    asm volatile("global_load_async_to_lds_b64 %0, %1, off"
                 :: "v"(lds_off), "v"(src) : "memory");
}
__device__ __forceinline__ void async_copy_b32(void* lds_dst, const void* src) {
    const uint32_t lds_off = (uint32_t)(uintptr_t)lds_dst;
    asm volatile("global_load_async_to_lds_b32 %0, %1, off"
                 :: "v"(lds_off), "v"(src) : "memory");
}
__device__ __forceinline__ void wait_async0() {
    asm volatile("s_wait_asynccnt 0x0" ::: "memory");
}

// ---------------------------------------------------------------------------
// Zero a float buffer (stats accumulators before atomic accumulation)
// ---------------------------------------------------------------------------
__global__ void zero_kernel(float* __restrict__ p, int n) {
    int i = blockIdx.x * 256 + threadIdx.x;
    if (i < n) p[i] = 0.0f;
}

// ---------------------------------------------------------------------------
// kNN (K=3) over squared distances. pos2 staged through LDS in 24KB tiles
// via async global->LDS copies; branchless top-3 insertion network so the
// 2048-iteration inner loop is straight-line VALU (wave32 friendly).
// ---------------------------------------------------------------------------
__global__ __launch_bounds__(256) void knn_kernel(
    const float* __restrict__ pos1, const float* __restrict__ pos2,
    int* __restrict__ nidx, float* __restrict__ nwts)
{
    __shared__ float xs[TILE_M];
    __shared__ float ys[TILE_M];
    __shared__ float zs[TILE_M];

    const int n = blockIdx.x * 256 + threadIdx.x;
    const float px = pos1[n * 3 + 0];
    const float py = pos1[n * 3 + 1];
    const float pz = pos1[n * 3 + 2];

    float d0 = 3.4e38f, d1 = 3.4e38f, d2 = 3.4e38f;
    int   i0 = 0, i1 = 0, i2 = 0;

    for (int t = 0; t < M_PTS / TILE_M; ++t) {
        __syncthreads();                       // previous tile fully consumed
        for (int p = threadIdx.x; p < TILE_M; p += 256) {
            const int g = t * TILE_M + p;
            async_copy_b32(&xs[p], pos2 + g * 3 + 0);
            async_copy_b32(&ys[p], pos2 + g * 3 + 1);
            async_copy_b32(&zs[p], pos2 + g * 3 + 2);
        }
        wait_async0();                         // this wave's copies landed
        __syncthreads();                       // all waves' copies landed
        const int base = t * TILE_M;
        #pragma unroll 4
        for (int p = 0; p < TILE_M; ++p) {     // LDS broadcast reads
            const float dx = px - xs[p];
            const float dy = py - ys[p];
            const float dz = pz - zs[p];
            const float d  = fmaf(dx, dx, fmaf(dy, dy, dz * dz));
            const int   j  = base + p;
            // branchless 3-deep insertion (compare/select network)
            const bool lt0 = d < d0, lt1 = d < d1, lt2 = d < d2;
            d2 = lt1 ? d1 : (lt2 ? d : d2);
            i2 = lt1 ? i1 : (lt2 ? j : i2);
            d1 = lt0 ? d0 : (lt1 ? d : d1);
            i1 = lt0 ? i0 : (lt1 ? j : i1);
            d0 = lt0 ? d  : d0;
            i0 = lt0 ? j  : i0;
        }
    }

    float w0 = 1.0f / (d0 + EPS_D);
    float w1 = 1.0f / (d1 + EPS_D);
    float w2 = 1.0f / (d2 + EPS_D);
    const float inv = 1.0f / (w0 + w1 + w2);
    nidx[n * 3 + 0] = i0; nidx[n * 3 + 1] = i1; nidx[n * 3 + 2] = i2;
    nwts[n * 3 + 0] = w0 * inv; nwts[n * 3 + 1] = w1 * inv; nwts[n * 3 + 2] = w2 * inv;
}

// ---------------------------------------------------------------------------
// Weighted gather-interpolation: one block (128 threads) per output row.
// feat2 (4MB) is L2-resident, so gathers are cheap; stores are coalesced.
// ---------------------------------------------------------------------------
__global__ __launch_bounds__(128) void interp_kernel(
    const float* __restrict__ feat2, const int* __restrict__ nidx,
    const float* __restrict__ nwts, float* __restrict__ xout)
{
    const int n = blockIdx.x;
    const int c = threadIdx.x;
    const int i0 = nidx[n * 3 + 0], i1 = nidx[n * 3 + 1], i2 = nidx[n * 3 + 2];
    const float w0 = nwts[n * 3 + 0], w1 = nwts[n * 3 + 1], w2 = nwts[n * 3 + 2];
    xout[n * C_DIM + c] = w0 * feat2[i0 * C_DIM + c]
                        + w1 * feat2[i1 * C_DIM + c]
                        + w2 * feat2[i2 * C_DIM + c];
}

// ---------------------------------------------------------------------------
// GEMM y = act(x) @ W^T + b with native f32 WMMA 16x16x4.
//   - HAS_BN (template): fused relu(x*scale+shift) on the input, scale/shift
//     read from LDS broadcast loads -> straight-line inner loop.
//   - W async-preloaded into LDS pre-swizzled into B-fragment layout so the
//     k-loop issues conflict-free ds_load_b64s.
//   - epilogue accumulates per-column sum / sum-of-squares for this layer's
//     BatchNorm via in-register reduction + shfl_xor + global atomics.
// One wave = 16-row strip x 128 output columns (8 WMMA tiles, 64 acc VGPRs).
// Block = 8 waves = 128 rows. Grid = 256 blocks (N=32768).
// Fragment layout (f32, wave32):
//   A 16x4 : lanes0-15 M=lane (K=k0,k0+1 in v.x,v.y); lanes16-31 K=k0+2,k0+3
//   B 4x16 : lanes0-15 N=lane (K=k0,k0+1);            lanes16-31 K=k0+2,k0+3
//   C/D    : VGPR r, lanes0-15 -> (M=r, N=lane); lanes16-31 -> (M=8+r)
// ---------------------------------------------------------------------------
template <bool HAS_BN>
__global__ __launch_bounds__(256) void gemm_bn_kernel(
    const float* __restrict__ X, const float* __restrict__ W,
    const float* __restrict__ bias,
    const float* __restrict__ scaleIn, const float* __restrict__ shiftIn,
    float* __restrict__ Y,
    float* __restrict__ colSum, float* __restrict__ colSq)
{
    __shared__ v2f wlds[8 * 32 * 32];            // [j][kq][lane] fragments, 64KB
    __shared__ __align__(8) float slds[C_DIM];   // BN scale
    __shared__ __align__(8) float hlds[C_DIM];   // BN shift

    const int tid = threadIdx.x;

    // Async preload of W (row-major [out=128][in=128]) into swizzled
    // B-fragment layout: memory -> LDS directly, ASYNCcnt-tracked.
    for (int s = tid; s < 8192; s += 256) {
        const int lane = s & 31;
        const int kq   = (s >> 5) & 31;
        const int j    = s >> 10;
        const int row  = j * 16 + (lane & 15);          // output feature (N)
        const int kb   = kq * 4 + (lane >> 4) * 2;      // K base for this half
        async_copy_b64(&wlds[s], W + row * C_DIM + kb);
    }
    if (HAS_BN && tid < C_DIM) {
        slds[tid] = scaleIn[tid];
        hlds[tid] = shiftIn[tid];
    }
    wait_async0();
    __syncthreads();

    const int lane  = tid & 31;
    const int wave  = tid >> 5;
    const int l15   = lane & 15;
    const int khalf = lane >> 4;
    const int row0  = blockIdx.x * 128 + wave * 16;     // strip base row
    const float* xrow = X + (row0 + l15) * C_DIM;

    v8f acc[8];
    #pragma unroll
    for (int j = 0; j < 8; ++j) {
        const float bv = bias[j * 16 + l15];            // column N = 16j + l15
        #pragma unroll
        for (int r = 0; r < 8; ++r) acc[j][r] = bv;
    }

    for (int kq = 0; kq < 32; ++kq) {
        const int kb = kq * 4 + khalf * 2;
        v2f a = *(const v2f*)(xrow + kb);
        if (HAS_BN) {   // fused BN + ReLU of previous layer (LDS broadcast)
            const v2f sc = *(const v2f*)(slds + kb);
            const v2f sh = *(const v2f*)(hlds + kb);
            a.x = fmaxf(fmaf(a.x, sc.x, sh.x), 0.0f);
            a.y = fmaxf(fmaf(a.y, sc.y, sh.y), 0.0f);
        }
        const v2f* wk = &wlds[kq * 32 + lane];
        #pragma unroll
        for (int j = 0; j < 8; ++j) {
            const v2f b = wk[j * 1024];
            acc[j] = __builtin_amdgcn_wmma_f32_16x16x4_f32(
                false, a, false, b, (short)0, acc[j], false, false);
        }
    }

    // Store y and accumulate per-column batch statistics.
    #pragma unroll
    for (int j = 0; j < 8; ++j) {
        float s = 0.0f, q = 0.0f;
        #pragma unroll
        for (int r = 0; r < 8; ++r) {
            const float v = acc[j][r];
            Y[(row0 + r + 8 * khalf) * C_DIM + j * 16 + l15] = v;
            s += v;
            q += v * v;
        }
        s += __shfl_xor(s, 16, 32);   // fold lane halves: rows 0-7 + rows 8-15
        q += __shfl_xor(q, 16, 32);
        if (khalf == 0) {
            atomicAdd(&colSum[j * 16 + l15], s);
            atomicAdd(&colSq [j * 16 + l15], q);
        }
    }
}

// ---------------------------------------------------------------------------
// Turn column sums into BN scale/shift:  relu((y-m)*rsqrt(v+eps)*g + be)
//   => relu(y*scale + shift),  scale = g*rsqrt(var+eps), shift = be - m*scale
// ---------------------------------------------------------------------------
__global__ __launch_bounds__(128) void finalize_kernel(
    const float* __restrict__ colSum, const float* __restrict__ colSq,
    const float* __restrict__ g, const float* __restrict__ be,
    float* __restrict__ scale, float* __restrict__ shift)
{
    const int c = threadIdx.x;
    const float invN = 1.0f / (float)N_PTS;
    const float mean = colSum[c] * invN;
    const float var  = colSq[c] * invN - mean * mean;   // biased var (ddof=0)
    const float s    = g[c] * rsqrtf(var + EPS_BN);
    scale[c] = s;
    shift[c] = be[c] - mean * s;
}

// ---------------------------------------------------------------------------
// Final BN+ReLU applied to last layer's pre-activation -> d_out.
// ---------------------------------------------------------------------------
__global__ __launch_bounds__(256) void bn_relu_kernel(
    const float* __restrict__ Yin, const float* __restrict__ scale,
    const float* __restrict__ shift, float* __restrict__ out)
{
    const int i = blockIdx.x * 256 + threadIdx.x;
    const int c = i & (C_DIM - 1);
    out[i] = fmaxf(Yin[i] * scale[c] + shift[c], 0.0f);
}

// ---------------------------------------------------------------------------
extern "C" void kernel_launch(void* const* d_in, const int* in_sizes, int n_in,
                              void* d_out, int out_size, void* d_ws, size_t ws_size,
                              hipStream_t stream)
{
    const float* pos1  = (const float*)d_in[0];
    const float* pos2  = (const float*)d_in[1];
    const float* feat2 = (const float*)d_in[2];
    const float* Wl[3]  = { (const float*)d_in[3],  (const float*)d_in[7],  (const float*)d_in[11] };
    const float* bl[3]  = { (const float*)d_in[4],  (const float*)d_in[8],  (const float*)d_in[12] };
    const float* gl[3]  = { (const float*)d_in[5],  (const float*)d_in[9],  (const float*)d_in[13] };
    const float* bel[3] = { (const float*)d_in[6],  (const float*)d_in[10], (const float*)d_in[14] };

    // Workspace layout
    char* ws = (char*)d_ws;
    const size_t featBytes = (size_t)N_PTS * C_DIM * sizeof(float);   // 16 MB
    float* bufA = (float*)ws;                       // x0 / y1
    float* bufB = (float*)(ws + featBytes);         // y0 / y2
    int*   nidx = (int*)  (ws + 2 * featBytes);
    float* nwts = (float*)(ws + 2 * featBytes + (size_t)N_PTS * 3 * sizeof(int));
    float* stats = (float*)(ws + 2 * featBytes + (size_t)N_PTS * 3 * (sizeof(int) + sizeof(float)));
    // stats: [sum0|sq0|sum1|sq1|sum2|sq2] then [scale0|shift0|...|scale2|shift2]
    float* colSum[3] = { stats + 0 * C_DIM, stats + 2 * C_DIM, stats + 4 * C_DIM };
    float* colSq [3] = { stats + 1 * C_DIM, stats + 3 * C_DIM, stats + 5 * C_DIM };
    float* scaleL[3] = { stats + 6 * C_DIM, stats + 8 * C_DIM, stats + 10 * C_DIM };
    float* shiftL[3] = { stats + 7 * C_DIM, stats + 9 * C_DIM, stats + 11 * C_DIM };

    // 0) zero the 6 stat accumulators (768 floats)
    zero_kernel<<<3, 256, 0, stream>>>(stats, 6 * C_DIM);

    // 1) kNN + weights
    knn_kernel<<<N_PTS / 256, 256, 0, stream>>>(pos1, pos2, nidx, nwts);

    // 2) interpolation -> bufA
    interp_kernel<<<N_PTS, C_DIM, 0, stream>>>(feat2, nidx, nwts, bufA);

    // 3) layer 0: y0 = x0 @ W0^T + b0 (+stats)        bufA -> bufB
    gemm_bn_kernel<false><<<N_PTS / 128, 256, 0, stream>>>(bufA, Wl[0], bl[0],
        nullptr, nullptr, bufB, colSum[0], colSq[0]);
    finalize_kernel<<<1, C_DIM, 0, stream>>>(colSum[0], colSq[0], gl[0], bel[0],
        scaleL[0], shiftL[0]);

    // 4) layer 1: y1 = relu(bn(y0)) @ W1^T + b1       bufB -> bufA
    gemm_bn_kernel<true><<<N_PTS / 128, 256, 0, stream>>>(bufB, Wl[1], bl[1],
        scaleL[0], shiftL[0], bufA, colSum[1], colSq[1]);
    finalize_kernel<<<1, C_DIM, 0, stream>>>(colSum[1], colSq[1], gl[1], bel[1],
        scaleL[1], shiftL[1]);

    // 5) layer 2: y2 = relu(bn(y1)) @ W2^T + b2       bufA -> bufB
    gemm_bn_kernel<true><<<N_PTS / 128, 256, 0, stream>>>(bufA, Wl[2], bl[2],
        scaleL[1], shiftL[1], bufB, colSum[2], colSq[2]);
    finalize_kernel<<<1, C_DIM, 0, stream>>>(colSum[2], colSq[2], gl[2], bel[2],
        scaleL[2], shiftL[2]);

    // 6) final BN + ReLU -> d_out
    bn_relu_kernel<<<(N_PTS * C_DIM) / 256, 256, 0, stream>>>(bufB,
        scaleL[2], shiftL[2], (float*)d_out);
}